// NMS_22273700397555
// MI455X (gfx1250) — compile-verified
//
#include <hip/hip_runtime.h>
#include <hip/hip_bf16.h>

#define NT   1024
#define NMAX 4096

#if defined(__has_builtin)
#if __has_builtin(__builtin_amdgcn_global_load_async_to_lds_b128) && \
    __has_builtin(__builtin_amdgcn_s_wait_asynccnt)
#define USE_ASYNC_LDS 1
#endif
#endif

typedef int v4i __attribute__((ext_vector_type(4)));
typedef __attribute__((address_space(1))) v4i* gv4i_ptr;  // global (AS1)
typedef __attribute__((address_space(3))) v4i* lv4i_ptr;  // LDS    (AS3)

__global__ __launch_bounds__(NT) void nms_fused_kernel(
    const float* __restrict__ boxes,   // N x 4 (x, y, w, h)
    const float* __restrict__ scores,  // N
    float* __restrict__ out,           // N x 5
    int n) {
  __shared__ float4        s_stage[NMAX];  // raw boxes, original order (64 KB)
  __shared__ float4        s_bb[NMAX];     // sorted (x, y, x2, y2)      (64 KB)
  __shared__ float         s_score[NMAX];  // 16 KB
  __shared__ int           s_idx[NMAX];    // 16 KB
  __shared__ float         s_area[NMAX];   // 16 KB
  __shared__ unsigned char s_keep[NMAX];   //  4 KB

  const int tid = threadIdx.x;

  int npad = 1;
  while (npad < n) npad <<= 1;  // n <= 4096 in this problem

  // ---- Stage raw boxes into LDS. Async DMA overlaps with the sort below. ----
#if USE_ASYNC_LDS
  for (int i = tid; i < n; i += NT) {
    __builtin_amdgcn_global_load_async_to_lds_b128(
        (gv4i_ptr)(v4i*)(void*)(boxes + 4 * i),
        (lv4i_ptr)(v4i*)(void*)&s_stage[i],
        0, 0);
  }
#else
  for (int i = tid; i < n; i += NT) {
    s_stage[i] = ((const float4*)boxes)[i];
  }
#endif

  // ---- Load scores / init indices (pad with -inf so padding sorts last). ----
  for (int i = tid; i < npad; i += NT) {
    s_score[i] = (i < n) ? scores[i] : -3.402823466e38f;
    s_idx[i]   = i;
  }

  // ---- In-LDS bitonic sort, descending, stable via original-index tiebreak
  //      (matches jnp.argsort(-scores) exactly, including ties). ----
  for (int k = 2; k <= npad; k <<= 1) {
    for (int j = k >> 1; j > 0; j >>= 1) {
      __syncthreads();
      for (int i = tid; i < npad; i += NT) {
        int ixj = i ^ j;
        if (ixj > i) {
          float a = s_score[i], b = s_score[ixj];
          int   ia = s_idx[i],  ib = s_idx[ixj];
          // "a precedes b" in the desired global (descending, stable) order
          bool aFirst   = (a > b) || ((a == b) && (ia < ib));
          bool wantDesc = ((i & k) == 0);
          if (wantDesc ? !aFirst : aFirst) {
            s_score[i] = b; s_score[ixj] = a;
            s_idx[i]   = ib; s_idx[ixj]  = ia;
          }
        }
      }
    }
  }

#if USE_ASYNC_LDS
  __builtin_amdgcn_s_wait_asynccnt(0);  // box staging must be complete now
#endif
  __syncthreads();

  // ---- Gather boxes into sorted order (LDS->LDS), precompute x2,y2,area. ----
  for (int i = tid; i < npad; i += NT) {
    if (i < n) {
      float4 b = s_stage[s_idx[i]];
      s_bb[i]   = make_float4(b.x, b.y, b.x + b.z, b.y + b.w);
      s_area[i] = b.z * b.w;
      s_keep[i] = (s_score[i] > 0.3f) ? (unsigned char)1 : (unsigned char)0;
    } else {
      s_bb[i]   = make_float4(0.f, 0.f, 0.f, 0.f);
      s_area[i] = 0.f;
      s_keep[i] = 0;
    }
  }
  __syncthreads();

  // ---- Serial NMS scan: row i (if kept) suppresses all j > i with IoU > 0.5.
  //      Inner loop parallel over 1024 threads; barrier only after kept rows
  //      (suppressed rows perform no LDS writes). Branch is uniform: every
  //      thread reads the same LDS byte after the last barrier. ----
  volatile unsigned char* vkeep = s_keep;
  for (int i = 0; i < n; ++i) {
    if (vkeep[i]) {
      float4 bi = s_bb[i];
      float  ai = s_area[i];
      for (int j = i + 1 + tid; j < n; j += NT) {
        if (vkeep[j]) {
          float4 bj = s_bb[j];
          float ax = fmaxf(bi.x, bj.x);
          float ay = fmaxf(bi.y, bj.y);
          float bx = fminf(bi.z, bj.z);
          float by = fminf(bi.w, bj.w);
          float inter = fmaxf(bx - ax, 0.f) * fmaxf(by - ay, 0.f);
          float uni   = fmaxf(ai + s_area[j] - inter, 1e-6f);
          float iou   = inter / uni;          // same form as reference
          if (iou > 0.5f) vkeep[j] = 0;
        }
      }
      __syncthreads();
    }
  }

  // ---- Emit [s*keep, x*keep, y*keep, w*keep, h*keep] per sorted row. ----
  for (int i = tid; i < n; i += NT) {
    float  kf = s_keep[i] ? 1.0f : 0.0f;
    float4 b  = s_stage[s_idx[i]];   // original (x, y, w, h)
    out[i * 5 + 0] = s_score[i] * kf;
    out[i * 5 + 1] = b.x * kf;
    out[i * 5 + 2] = b.y * kf;
    out[i * 5 + 3] = b.z * kf;
    out[i * 5 + 4] = b.w * kf;
  }
}

extern "C" void kernel_launch(void* const* d_in, const int* in_sizes, int n_in,
                              void* d_out, int out_size, void* d_ws, size_t ws_size,
                              hipStream_t stream) {
  (void)n_in; (void)out_size; (void)d_ws; (void)ws_size;
  const float* boxes  = (const float*)d_in[0];
  const float* scores = (const float*)d_in[1];
  float*       out    = (float*)d_out;
  int n = in_sizes[1];  // number of boxes (4096)

  nms_fused_kernel<<<1, NT, 0, stream>>>(boxes, scores, out, n);
}